// SageConv_51084341018869
// MI455X (gfx1250) — compile-verified
//
#include <hip/hip_runtime.h>
#include <math.h>

#define N_NODES 50000
#define D 128

typedef __attribute__((ext_vector_type(2))) float v2f;
typedef __attribute__((ext_vector_type(8))) float v8f;

// ---------------------------------------------------------------------------
// Kernel 1: edge scatter (segment-sum + degree) via native f32 L2 atomics.
// One wave per edge: 32 lanes x float4 = 128 channels.
// ---------------------------------------------------------------------------
__global__ __launch_bounds__(256)
void sage_scatter(const float* __restrict__ h,
                  const int* __restrict__ row,
                  const int* __restrict__ col,
                  float* __restrict__ summed,
                  float* __restrict__ deg,
                  int nEdges)
{
    const int lane  = threadIdx.x & 31;
    const int gwave = (blockIdx.x * blockDim.x + threadIdx.x) >> 5;
    const int nwave = (gridDim.x * blockDim.x) >> 5;

    for (int e = gwave; e < nEdges; e += nwave) {
        const int r = row[e];
        const int c = col[e];
        const float4 v = *(const float4*)(h + (size_t)c * D + lane * 4);
        float* dst = summed + (size_t)r * D + lane * 4;
        __hip_atomic_fetch_add(dst + 0, v.x, __ATOMIC_RELAXED, __HIP_MEMORY_SCOPE_AGENT);
        __hip_atomic_fetch_add(dst + 1, v.y, __ATOMIC_RELAXED, __HIP_MEMORY_SCOPE_AGENT);
        __hip_atomic_fetch_add(dst + 2, v.z, __ATOMIC_RELAXED, __HIP_MEMORY_SCOPE_AGENT);
        __hip_atomic_fetch_add(dst + 3, v.w, __ATOMIC_RELAXED, __HIP_MEMORY_SCOPE_AGENT);
        if (lane == 0) {
            __hip_atomic_fetch_add(deg + r, 1.0f, __ATOMIC_RELAXED, __HIP_MEMORY_SCOPE_AGENT);
        }
    }
}

// ---------------------------------------------------------------------------
// Kernel 2: out = normalize(h_target @ W1 + (summed/deg) @ W2 + b2)
// One wave per 16-node tile; fp32 WMMA 16x16x4; all 8 N-tiles (128 cols)
// accumulated in registers so row L2-norm is computed in-register.
//
// WMMA f32 layouts (wave32):
//   A 16x4:  lane<16 -> M=lane, vgpr{0,1}=K{0,1};  lane>=16 -> M=lane-16, K{2,3}
//   B 4x16:  lane<16 -> N=lane, vgpr{0,1}=K{0,1};  lane>=16 -> N=lane-16, K{2,3}
//   C 16x16: vgpr r, lane<16 -> (M=r,   N=lane)
//            vgpr r, lane>=16-> (M=r+8, N=lane-16)
// ---------------------------------------------------------------------------
__global__ __launch_bounds__(256)
void sage_gemm_norm(const float* __restrict__ h_target,
                    const float* __restrict__ summed,
                    const float* __restrict__ deg,
                    const float* __restrict__ W1,
                    const float* __restrict__ W2,
                    const float* __restrict__ bias,
                    float* __restrict__ out)
{
    const int lane = threadIdx.x & 31;
    const int wave = threadIdx.x >> 5;
    const int m0   = (blockIdx.x * (blockDim.x >> 5) + wave) * 16;
    if (m0 >= N_NODES) return;               // wave-uniform: EXEC stays all-1s

    const int half  = lane >> 4;             // 0: K 0..1 / M r ; 1: K 2..3 / M r+8
    const int mr    = lane & 15;
    const int myrow = m0 + mr;

    const float dgv  = deg[myrow];
    const float invd = 1.0f / fmaxf(dgv, 1.0f);
    const float* At  = h_target + (size_t)myrow * D;   // A1 rows
    const float* As  = summed   + (size_t)myrow * D;   // A2 rows (pre-mean)

    // Accumulators: 8 N-tiles of 16x16, bias folded into init.
    v8f acc[8];
#pragma unroll
    for (int nt = 0; nt < 8; ++nt) {
        const float bv = bias[nt * 16 + mr];
#pragma unroll
        for (int i = 0; i < 8; ++i) acc[nt][i] = bv;
    }

    // K = 128 in steps of 4; two WMMAs (W1 path + W2 path) per (k, n-tile).
    for (int kk = 0; kk < 32; ++kk) {
        const int k0 = kk * 4 + half * 2;
        v2f a1 = *(const v2f*)(At + k0);
        v2f a2 = *(const v2f*)(As + k0);
        a2[0] *= invd;
        a2[1] *= invd;   // fused mean aggregation

        const float* w1p = W1 + (size_t)k0 * D + mr;
        const float* w2p = W2 + (size_t)k0 * D + mr;
#pragma unroll
        for (int nt = 0; nt < 8; ++nt) {
            const int nc = nt * 16;
            v2f b1, b2v;
            b1[0]  = w1p[nc];     b1[1]  = w1p[D + nc];
            b2v[0] = w2p[nc];     b2v[1] = w2p[D + nc];
            acc[nt] = __builtin_amdgcn_wmma_f32_16x16x4_f32(
                false, a1, false, b1,  (short)0, acc[nt], false, false);
            acc[nt] = __builtin_amdgcn_wmma_f32_16x16x4_f32(
                false, a2, false, b2v, (short)0, acc[nt], false, false);
        }
    }

    // Row L2 norm: sum of squares over the 128 outputs of each row.
    // Row M = r + 8*half lives in acc[*][r] across the 16 lanes of this half;
    // xor-shuffle masks 1,2,4,8 reduce within each 16-lane group.
    float scl[8];
#pragma unroll
    for (int r = 0; r < 8; ++r) {
        float s = 0.0f;
#pragma unroll
        for (int nt = 0; nt < 8; ++nt) s += acc[nt][r] * acc[nt][r];
        s += __shfl_xor(s, 1, 32);
        s += __shfl_xor(s, 2, 32);
        s += __shfl_xor(s, 4, 32);
        s += __shfl_xor(s, 8, 32);
        scl[r] = 1.0f / fmaxf(sqrtf(s), 1e-12f);
    }

    // Store normalized outputs (lanes contiguous in N -> coalesced 64B/half).
#pragma unroll
    for (int nt = 0; nt < 8; ++nt) {
#pragma unroll
        for (int r = 0; r < 8; ++r) {
            out[(size_t)(m0 + r + 8 * half) * D + nt * 16 + mr] = acc[nt][r] * scl[r];
        }
    }
}

// ---------------------------------------------------------------------------
extern "C" void kernel_launch(void* const* d_in, const int* in_sizes, int n_in,
                              void* d_out, int out_size, void* d_ws, size_t ws_size,
                              hipStream_t stream) {
    const float* h        = (const float*)d_in[0];
    const float* h_target = (const float*)d_in[1];
    const int*   row      = (const int*)d_in[2];
    const int*   col      = (const int*)d_in[3];
    const float* W1       = (const float*)d_in[4];
    const float* W2       = (const float*)d_in[5];
    const float* b2       = (const float*)d_in[6];
    float*       out      = (float*)d_out;
    const int    nEdges   = in_sizes[2];

    float* summed = (float*)d_ws;                       // [N_NODES, D]
    float* deg    = summed + (size_t)N_NODES * D;       // [N_NODES]
    const size_t zbytes = ((size_t)N_NODES * D + N_NODES) * sizeof(float);

    hipMemsetAsync(d_ws, 0, zbytes, stream);

    sage_scatter<<<2048, 256, 0, stream>>>(h, row, col, summed, deg, nEdges);

    const int waves_per_block = 256 / 32;
    const int nodes_per_block = waves_per_block * 16;
    const int blocks = (N_NODES + nodes_per_block - 1) / nodes_per_block;
    sage_gemm_norm<<<blocks, 256, 0, stream>>>(h_target, summed, deg, W1, W2, b2, out);
}